// GraphAutoEncoder_88794153877682
// MI455X (gfx1250) — compile-verified
//
#include <hip/hip_runtime.h>

namespace {
constexpr int kN = 50000;   // nodes
constexpr int kD = 128;     // feature dim
}

typedef __attribute__((ext_vector_type(2))) float v2f;
typedef __attribute__((ext_vector_type(8))) float v8f;

// ---------------------------------------------------------------------------
// residual copy: out = in  (scatter kernel then accumulates edge messages
// on top, giving out = in + segment_sum(msg))
// ---------------------------------------------------------------------------
__global__ __launch_bounds__(256)
void copy_vec4(const float4* __restrict__ in, float4* __restrict__ out, int n4) {
  int i = blockIdx.x * blockDim.x + threadIdx.x;
  int stride = gridDim.x * blockDim.x;
  for (; i < n4; i += stride) out[i] = in[i];
}

// ---------------------------------------------------------------------------
// one wave per edge: 32 lanes x float4 = 128 features.
// out[dst] += w * feat[src]   via native f32 global atomics.
// ---------------------------------------------------------------------------
__global__ __launch_bounds__(256)
void edge_scatter(const float* __restrict__ feat,
                  const int* __restrict__ src,
                  const int* __restrict__ dst,
                  const float* __restrict__ ea,
                  float* __restrict__ out, int nedge) {
  int e = (int)((blockIdx.x * (unsigned)blockDim.x + threadIdx.x) >> 5);
  if (e >= nedge) return;
  int lane = threadIdx.x & 31;
  int s = src[e];
  int d = dst[e];
  float w = ea[e];
  const float4 v = *(const float4*)(feat + (size_t)s * kD + lane * 4);
  float* o = out + (size_t)d * kD + lane * 4;
  unsafeAtomicAdd(o + 0, w * v.x);
  unsafeAtomicAdd(o + 1, w * v.y);
  unsafeAtomicAdd(o + 2, w * v.z);
  unsafeAtomicAdd(o + 3, w * v.w);
}

// ---------------------------------------------------------------------------
// C[M,Nout] = act(A[M,K] @ B[K,Nout] + bias), fp32 WMMA 16x16x4.
// One wave computes one 16x16 output tile; per-lane operand layout follows
// ISA 7.12.2 (lanes 16-31 carry K+2/K+3 for A/B, rows M+8 for C/D).
// ---------------------------------------------------------------------------
__global__ __launch_bounds__(128)
void gemm_bias_act_wmma(const float* __restrict__ A, const float* __restrict__ B,
                        const float* __restrict__ bias, float* __restrict__ C,
                        int M, int K, int Nout, int relu) {
  const int ntn   = Nout >> 4;
  const int total = (M >> 4) * ntn;
  const int t = (int)(blockIdx.x * (blockDim.x >> 5) + (threadIdx.x >> 5)); // wave-uniform
  if (t >= total) return;

  const int m0   = (t / ntn) << 4;
  const int n0   = (t % ntn) << 4;
  const int lane = threadIdx.x & 31;
  const int half = lane >> 4;   // 0: K,K+1 / rows M..M+7 ; 1: K+2,K+3 / rows M+8..M+15
  const int r    = lane & 15;

  const float* Arow = A + (size_t)(m0 + r) * K + 2 * half;  // contiguous float2 per step
  const float* Bcol = B + n0 + r;

  v8f acc = {};
  #pragma unroll 4
  for (int k = 0; k < K; k += 4) {
    v2f a, b;
    a.x = Arow[k];
    a.y = Arow[k + 1];
    const float* bp = Bcol + (size_t)(k + 2 * half) * Nout;
    b.x = bp[0];
    b.y = bp[Nout];
    // D = A(16x4) * B(4x16) + C, full fp32
    acc = __builtin_amdgcn_wmma_f32_16x16x4_f32(
        /*neg_a=*/false, a, /*neg_b=*/false, b,
        /*c_mod=*/(short)0, acc, /*reuse_a=*/false, /*reuse_b=*/false);
  }

  const float bv = bias[n0 + r];
  float* Crow = C + (size_t)m0 * Nout + n0 + r;
  #pragma unroll
  for (int v = 0; v < 8; ++v) {
    float x = acc[v] + bv;
    if (relu) x = fmaxf(x, 0.0f);
    Crow[(size_t)(v + 8 * half) * Nout] = x;
  }
}

// ---------------------------------------------------------------------------
extern "C" void kernel_launch(void* const* d_in, const int* in_sizes, int n_in,
                              void* d_out, int out_size, void* d_ws, size_t ws_size,
                              hipStream_t stream) {
  const float* x   = (const float*)d_in[0];
  const int*   ei  = (const int*)d_in[1];   // [2, E]
  const float* ea  = (const float*)d_in[2];
  const float* We0 = (const float*)d_in[3];
  const float* be0 = (const float*)d_in[4];
  const float* We1 = (const float*)d_in[5];
  const float* be1 = (const float*)d_in[6];
  const float* We2 = (const float*)d_in[7];
  const float* be2 = (const float*)d_in[8];
  const float* Wd0 = (const float*)d_in[9];
  const float* bd0 = (const float*)d_in[10];
  const float* Wd1 = (const float*)d_in[11];
  const float* bd1 = (const float*)d_in[12];
  const float* Wd2 = (const float*)d_in[13];
  const float* bd2 = (const float*)d_in[14];

  const int E = in_sizes[2];
  const int* srcIdx = ei;
  const int* dstIdx = ei + E;

  float* out_agg = (float*)d_out;                       // [N,128]
  float* out_dec = out_agg + (size_t)kN * kD;           // [N,128]

  char* ws = (char*)d_ws;
  float* R1 = (float*)ws;                                        // N*512 f32
  float* R2 = (float*)(ws + (size_t)kN * 512 * sizeof(float));   // N*512 f32
  float* R3 = (float*)(ws + (size_t)kN * 512 * sizeof(float)*2); // N*64  f32

  const int n4 = kN * kD / 4;
  const int scatBlocks = (E * 32 + 255) / 256;

  // --- 3 aggregation rounds: out = in + scatter_add(w * in[src]) ---
  copy_vec4  <<<512, 256, 0, stream>>>((const float4*)x,  (float4*)R1, n4);
  edge_scatter<<<scatBlocks, 256, 0, stream>>>(x,  srcIdx, dstIdx, ea, R1, E);
  copy_vec4  <<<512, 256, 0, stream>>>((const float4*)R1, (float4*)R2, n4);
  edge_scatter<<<scatBlocks, 256, 0, stream>>>(R1, srcIdx, dstIdx, ea, R2, E);
  copy_vec4  <<<512, 256, 0, stream>>>((const float4*)R2, (float4*)out_agg, n4);
  edge_scatter<<<scatBlocks, 256, 0, stream>>>(R2, srcIdx, dstIdx, ea, out_agg, E);

  // --- encoder / decoder MLPs, fp32 WMMA GEMMs ---
  auto gemm = [&](const float* A, const float* B, const float* bias, float* C,
                  int K, int Nout, int relu) {
    const int total  = (kN / 16) * (Nout / 16);
    const int blocks = (total + 3) / 4;   // 4 waves (16x16 tiles) per 128-thread block
    gemm_bias_act_wmma<<<blocks, 128, 0, stream>>>(A, B, bias, C, kN, K, Nout, relu);
  };

  gemm(out_agg, We0, be0, R1,      128, 512, 1);   // 128 -> 512, relu
  gemm(R1,      We1, be1, R2,      512, 256, 1);   // 512 -> 256, relu
  gemm(R2,      We2, be2, R3,      256,  64, 0);   // 256 ->  64
  gemm(R3,      Wd0, bd0, R1,       64, 256, 1);   //  64 -> 256, relu
  gemm(R1,      Wd1, bd1, R2,      256, 512, 1);   // 256 -> 512, relu
  gemm(R2,      Wd2, bd2, out_dec, 512, 128, 0);   // 512 -> 128
}